// DeepDendriticEncoder_65352222376340
// MI455X (gfx1250) — compile-verified
//
#include <hip/hip_runtime.h>
#include <math.h>

// ---------------------------------------------------------------------------
// DeepDendriticEncoder for MI455X (gfx1250, wave32)
//   h1 = relu(X @ W1.T)   X: Hankel windows of x, [N,256] x [256,128]
//   h2 = relu(h1 @ W2.T)  [N,128] x [128,64]
//   h3 = relu(h2 @ W3.T)  [N,64]  x [64,32]
//   I = 2*max(h3), winner = argmax(h3), closed-form LIF latency, global argmin
// All GEMMs via V_WMMA_F32_16X16X4_F32 (full fp32, matches reference math).
// Each wave keeps every N-tile accumulator live -> one A-fragment feeds
// 8/4/2 independent WMMA chains (back-to-back matrix issue, 8x less A traffic).
// ---------------------------------------------------------------------------

typedef __attribute__((ext_vector_type(2))) float v2f;
typedef __attribute__((ext_vector_type(8))) float v8f;

#define WWIN   256
#define K1     128
#define K2     64
#define K3     32
#define MTILE  128            // windows per block
#define NTHR   256            // 8 waves
#define P1     130            // h1/w2 row pitch (stride%64==2 -> no bank conflicts)
#define P2     66             // h2/w3 row pitch
#define P3     34             // h3 row pitch

__device__ __forceinline__ v8f wmma_f32(v2f a, v2f b, v8f c) {
  // D = A(16x4,f32) * B(4x16,f32) + C(16x16,f32)
  return __builtin_amdgcn_wmma_f32_16x16x4_f32(false, a, false, b, (short)0, c,
                                               false, false);
}

__global__ __launch_bounds__(NTHR) void dde_main(
    const float* __restrict__ x, const float* __restrict__ W1,
    const float* __restrict__ W2, const float* __restrict__ W3,
    float4* __restrict__ blockOut, int N, int T) {
  __shared__ float xseg[MTILE + WWIN];       // 384 floats
  __shared__ float w2s[K2 * P1];             // W2 padded
  __shared__ float w3s[K3 * P2];             // W3 padded
  __shared__ float h1s[MTILE * P1];
  __shared__ float h2s[MTILE * P2];
  __shared__ float h3s[MTILE * P3];
  __shared__ float r_time[MTILE];
  __shared__ float r_lat[MTILE];
  __shared__ int   r_idx[MTILE];
  __shared__ int   r_win[MTILE];

  const int tid  = threadIdx.x;
  const int wave = tid >> 5;
  const int lane = tid & 31;
  const int l16  = lane & 15;
  const int koff = (lane < 16) ? 0 : 2;      // K half per A/B VGPR layout
  const int gbase = blockIdx.x * MTILE;

  // ---- stage LDS: x segment, W2, W3 (coalesced) ----
  for (int i = tid; i < MTILE + WWIN; i += NTHR) {
    int gi = gbase + i;
    xseg[i] = (gi < T) ? x[gi] : 0.0f;
  }
  for (int i = tid; i < K2 * K1; i += NTHR)
    w2s[(i >> 7) * P1 + (i & 127)] = W2[i];
  for (int i = tid; i < K3 * K2; i += NTHR)
    w3s[(i >> 6) * P2 + (i & 63)] = W3[i];
  __syncthreads();

  const int mrow = (wave << 4) + l16;                   // A-fragment row (window)
  const int crow = (wave << 4) + ((lane < 16) ? 0 : 8); // C-fragment base row

  // ---- layer 1: h1[128][128] = relu(X @ W1.T), K=256, 8 N-tiles ----
  {
    v8f acc[8] = {};
    const float* ap = xseg + mrow + koff;               // Hankel A from LDS
    const float* bp = W1 + l16 * WWIN + koff;           // B tiles: +nt*16KB imm
#pragma unroll 2
    for (int k0 = 0; k0 < WWIN; k0 += 4) {
      v2f a; a.x = ap[k0]; a.y = ap[k0 + 1];
#pragma unroll
      for (int nt = 0; nt < 8; ++nt) {
        v2f b;
        b.x = bp[k0 + nt * 16 * WWIN];
        b.y = bp[k0 + nt * 16 * WWIN + 1];
        acc[nt] = wmma_f32(a, b, acc[nt]);
      }
    }
#pragma unroll
    for (int nt = 0; nt < 8; ++nt) {
      const int ccol = (nt << 4) + l16;
#pragma unroll
      for (int r = 0; r < 8; ++r) {
        float v = acc[nt][r];
        h1s[(crow + r) * P1 + ccol] = v > 0.0f ? v : 0.0f;
      }
    }
  }
  __syncthreads();

  // ---- layer 2: h2[128][64] = relu(h1 @ W2.T), K=128, 4 N-tiles ----
  {
    v8f acc[4] = {};
    const float* ap = &h1s[mrow * P1 + koff];
    const float* bp = &w2s[l16 * P1 + koff];
#pragma unroll 2
    for (int k0 = 0; k0 < K1; k0 += 4) {
      v2f a; a.x = ap[k0]; a.y = ap[k0 + 1];
#pragma unroll
      for (int nt = 0; nt < 4; ++nt) {
        v2f b;
        b.x = bp[k0 + nt * 16 * P1];
        b.y = bp[k0 + nt * 16 * P1 + 1];
        acc[nt] = wmma_f32(a, b, acc[nt]);
      }
    }
#pragma unroll
    for (int nt = 0; nt < 4; ++nt) {
      const int ccol = (nt << 4) + l16;
#pragma unroll
      for (int r = 0; r < 8; ++r) {
        float v = acc[nt][r];
        h2s[(crow + r) * P2 + ccol] = v > 0.0f ? v : 0.0f;
      }
    }
  }
  __syncthreads();

  // ---- layer 3: h3[128][32] = relu(h2 @ W3.T), K=64, 2 N-tiles ----
  {
    v8f acc[2] = {};
    const float* ap = &h2s[mrow * P2 + koff];
    const float* bp = &w3s[l16 * P2 + koff];
#pragma unroll
    for (int k0 = 0; k0 < K2; k0 += 4) {
      v2f a; a.x = ap[k0]; a.y = ap[k0 + 1];
#pragma unroll
      for (int nt = 0; nt < 2; ++nt) {
        v2f b;
        b.x = bp[k0 + nt * 16 * P2];
        b.y = bp[k0 + nt * 16 * P2 + 1];
        acc[nt] = wmma_f32(a, b, acc[nt]);
      }
    }
#pragma unroll
    for (int nt = 0; nt < 2; ++nt) {
      const int ccol = (nt << 4) + l16;
#pragma unroll
      for (int r = 0; r < 8; ++r) {
        float v = acc[nt][r];
        h3s[(crow + r) * P3 + ccol] = v > 0.0f ? v : 0.0f;
      }
    }
  }
  __syncthreads();

  // ---- soma: per-window max/argmax, LIF latency, block argmin ----
  if (tid < MTILE) {
    const float* hr = &h3s[tid * P3];
    float mx = hr[0];
    int arg = 0;
#pragma unroll
    for (int j = 1; j < K3; ++j) {
      float v = hr[j];
      if (v > mx) { mx = v; arg = j; }   // strict > keeps first max (argmax)
    }
    const float I = 2.0f * mx;
    const int gidx = gbase + tid;
    const bool valid = gidx < N;
    float lat = __builtin_inff();
    if (valid && I > 1.0f) {
      float safe = 1.0f - 1.0f / fmaxf(I, 1.0f + 1e-12f);
      float nst = ceilf(logf(safe) / logf(0.8f));   // DECAY = 0.8
      nst = fmaxf(nst, 1.0f);
      if (nst <= 200000.0f) lat = nst * 0.01f;      // DT = 0.01
    }
    r_time[tid] = valid ? (float)gidx + lat : __builtin_inff();
    r_lat[tid]  = lat;
    r_idx[tid]  = valid ? gidx : 0x7fffffff;
    r_win[tid]  = arg;
  }
  __syncthreads();
  if (tid == 0) {
    float bt = r_time[0], bl = r_lat[0];
    int bi = r_idx[0], bw = r_win[0];
    for (int j = 1; j < MTILE; ++j) {
      float t = r_time[j];
      int ix = r_idx[j];
      if (t < bt || (t == bt && ix < bi)) {   // first-index tie-break (inf==inf)
        bt = t; bl = r_lat[j]; bi = ix; bw = r_win[j];
      }
    }
    float4 res;
    res.x = bt;
    res.y = bl;
    res.z = __int_as_float(bi);
    res.w = __int_as_float(bw);
    blockOut[blockIdx.x] = res;
  }
}

__global__ __launch_bounds__(NTHR) void dde_reduce(
    const float4* __restrict__ blockOut, int nBlocks, float* __restrict__ out) {
  __shared__ float s_t[NTHR], s_l[NTHR];
  __shared__ int s_i[NTHR], s_w[NTHR];
  const int tid = threadIdx.x;
  float bt = __builtin_inff(), bl = __builtin_inff();
  int bi = 0x7fffffff, bw = 0;
  for (int j = tid; j < nBlocks; j += NTHR) {
    float4 r = blockOut[j];
    int ix = __float_as_int(r.z);
    if (r.x < bt || (r.x == bt && ix < bi)) {
      bt = r.x; bl = r.y; bi = ix; bw = __float_as_int(r.w);
    }
  }
  s_t[tid] = bt; s_l[tid] = bl; s_i[tid] = bi; s_w[tid] = bw;
  __syncthreads();
  if (tid == 0) {
    for (int j = 1; j < NTHR; ++j) {
      if (s_t[j] < bt || (s_t[j] == bt && s_i[j] < bi)) {
        bt = s_t[j]; bl = s_l[j]; bi = s_i[j]; bw = s_w[j];
      }
    }
    out[0] = (float)bi;   // best window index
    out[1] = (float)bw;   // winner unit
    out[2] = bl;          // latency
    out[3] = bt;          // abs_time
  }
}

extern "C" void kernel_launch(void* const* d_in, const int* in_sizes, int n_in,
                              void* d_out, int out_size, void* d_ws, size_t ws_size,
                              hipStream_t stream) {
  const float* x  = (const float*)d_in[0];   // [T]
  const float* W1 = (const float*)d_in[1];   // [128,256]
  const float* W2 = (const float*)d_in[2];   // [64,128]
  const float* W3 = (const float*)d_in[3];   // [32,64]
  const int T = in_sizes[0];
  const int N = T - WWIN + 1;
  const int nBlocks = (N + MTILE - 1) / MTILE;

  float4* blockOut = (float4*)d_ws;          // nBlocks * 16 bytes

  dde_main<<<nBlocks, NTHR, 0, stream>>>(x, W1, W2, W3, blockOut, N, T);
  dde_reduce<<<1, NTHR, 0, stream>>>(blockOut, nBlocks, (float*)d_out);
}